// QuAN8_1975684956565
// MI455X (gfx1250) — compile-verified
//
#include <hip/hip_runtime.h>
#include <math.h>

typedef float v2f __attribute__((ext_vector_type(2)));
typedef float v8f __attribute__((ext_vector_type(8)));

#if defined(__HIP_DEVICE_COMPILE__) && \
    !__has_builtin(__builtin_amdgcn_wmma_f32_16x16x4_f32)
#error "wmma_f32_16x16x4_f32 builtin not available on this device toolchain"
#endif

__device__ __forceinline__ v8f wmma4(v2f a, v2f b, v8f c) {
  // D = A(16x4 f32) * B(4x16 f32) + C(16x16 f32)
  return __builtin_amdgcn_wmma_f32_16x16x4_f32(false, a, false, b, (short)0, c,
                                               false, false);
}

__device__ __forceinline__ float wave_sum32(float v) {
  v += __shfl_xor(v, 1, 32);
  v += __shfl_xor(v, 2, 32);
  v += __shfl_xor(v, 4, 32);
  v += __shfl_xor(v, 8, 32);
  v += __shfl_xor(v, 16, 32);
  return v;
}

// ---------------------------------------------------------------------------
// Conv 2x2 (1->32ch on 4x4) + per-channel sum/sumsq accumulation (BN stats)
// ---------------------------------------------------------------------------
__global__ __launch_bounds__(256) void conv_stats_kernel(
    const float* __restrict__ X, const float* __restrict__ CW,
    float* __restrict__ Y, float* __restrict__ gsum, float* __restrict__ gsq) {
  const int tid = threadIdx.x;
  const int c = tid & 31;
  const int jn = tid >> 5;                 // 8 samples per block
  const int n = blockIdx.x * 8 + jn;       // n in [0, 32768)
  const float* x = X + (size_t)n * 16;
  const float* w = CW + c * 4;
  const float w0 = w[0], w1 = w[1], w2 = w[2], w3 = w[3];
  float* y = Y + (size_t)n * 288 + c * 9;
  float s = 0.f, s2 = 0.f;
#pragma unroll
  for (int i = 0; i < 3; ++i) {
#pragma unroll
    for (int j = 0; j < 3; ++j) {
      float v = w0 * x[i * 4 + j] + w1 * x[i * 4 + j + 1] +
                w2 * x[(i + 1) * 4 + j] + w3 * x[(i + 1) * 4 + j + 1];
      y[i * 3 + j] = v;
      s += v;
      s2 += v * v;
    }
  }
  __shared__ float rs[256], rq[256];
  rs[tid] = s;
  rq[tid] = s2;
  __syncthreads();
  if (tid < 32) {
    float a = 0.f, b = 0.f;
#pragma unroll
    for (int k = 0; k < 8; ++k) {
      a += rs[tid + 32 * k];
      b += rq[tid + 32 * k];
    }
    atomicAdd(&gsum[tid], a);
    atomicAdd(&gsq[tid], b);
  }
}

__global__ void bn_finalize_kernel(const float* __restrict__ gsum,
                                   const float* __restrict__ gsq,
                                   const float* __restrict__ g,
                                   const float* __restrict__ b,
                                   float* __restrict__ scale,
                                   float* __restrict__ shift) {
  const int c = threadIdx.x;  // 32 threads
  const float cnt = 32768.0f * 9.0f;
  float mean = gsum[c] / cnt;
  float var = gsq[c] / cnt - mean * mean;
  float rs = rsqrtf(var + 1e-5f);
  float sc = g[c] * rs;
  scale[c] = sc;
  shift[c] = b[c] - mean * sc;
}

__global__ __launch_bounds__(256) void bn_apply_kernel(
    const float* __restrict__ Y, const float* __restrict__ scale,
    const float* __restrict__ shift, float* __restrict__ Z) {
  const int idx = blockIdx.x * 256 + threadIdx.x;  // total 32768*288
  const int c = (idx % 288) / 9;
  Z[idx] = Y[idx] * scale[c] + shift[c];
}

// ---------------------------------------------------------------------------
// WMMA f32 GEMM: C[M,N] = A[M,K] @ W^T + bias  (W is N x K row-major)
// mode 0: C = acc + bias ; mode 1: C = R + sigmoid(acc + bias)
// Wave tile 32x64 (8 accumulators), block tile 256x64, grid (M/256, N/64).
// K loop is register double-buffered (ping/pong chunks of 4) so loads for the
// next chunk are in flight while WMMAs consume the current chunk.
// Requires K % 8 == 0 (K = 256 or 288 here).
// ---------------------------------------------------------------------------
__global__ __launch_bounds__(256) void gemm_wmma_kernel(
    const float* __restrict__ A, const float* __restrict__ W,
    const float* __restrict__ bias, const float* __restrict__ R,
    float* __restrict__ C, int M, int N, int K, int mode) {
  const int lane = threadIdx.x & 31;
  const int wv = threadIdx.x >> 5;
  const int lr = lane & 15;
  const int hi = lane >> 4;  // 0: K pair {0,1}, 1: K pair {2,3}
  const int m0 = blockIdx.x * 256 + wv * 32;
  const int n0 = blockIdx.y * 64;

  const float* Ar0 = A + (size_t)(m0 + lr) * K;
  const float* Ar1 = A + (size_t)(m0 + 16 + lr) * K;
  const float* Wr0 = W + (size_t)(n0 + 0 * 16 + lr) * K;
  const float* Wr1 = W + (size_t)(n0 + 1 * 16 + lr) * K;
  const float* Wr2 = W + (size_t)(n0 + 2 * 16 + lr) * K;
  const float* Wr3 = W + (size_t)(n0 + 3 * 16 + lr) * K;

  v8f acc[8];
#pragma unroll
  for (int i = 0; i < 8; ++i) acc[i] = (v8f){};

  // ping chunk (k = 0)
  v2f aA0, aA1, bA0, bA1, bA2, bA3;
  // pong chunk
  v2f aB0, aB1, bB0, bB1, bB2, bB3;

  {
    const int ko = 2 * hi;
    aA0 = *(const v2f*)(Ar0 + ko);
    aA1 = *(const v2f*)(Ar1 + ko);
    bA0 = *(const v2f*)(Wr0 + ko);
    bA1 = *(const v2f*)(Wr1 + ko);
    bA2 = *(const v2f*)(Wr2 + ko);
    bA3 = *(const v2f*)(Wr3 + ko);
  }

  for (int kb = 0; kb < K; kb += 8) {
    {  // prefetch pong chunk (kb + 4) while ping chunk computes
      const int ko = kb + 4 + 2 * hi;
      aB0 = *(const v2f*)(Ar0 + ko);
      aB1 = *(const v2f*)(Ar1 + ko);
      bB0 = *(const v2f*)(Wr0 + ko);
      bB1 = *(const v2f*)(Wr1 + ko);
      bB2 = *(const v2f*)(Wr2 + ko);
      bB3 = *(const v2f*)(Wr3 + ko);
    }
    acc[0] = wmma4(aA0, bA0, acc[0]);
    acc[1] = wmma4(aA1, bA0, acc[1]);
    acc[2] = wmma4(aA0, bA1, acc[2]);
    acc[3] = wmma4(aA1, bA1, acc[3]);
    acc[4] = wmma4(aA0, bA2, acc[4]);
    acc[5] = wmma4(aA1, bA2, acc[5]);
    acc[6] = wmma4(aA0, bA3, acc[6]);
    acc[7] = wmma4(aA1, bA3, acc[7]);
    if (kb + 8 < K) {  // prefetch next ping chunk (kb + 8)
      const int ko = kb + 8 + 2 * hi;
      aA0 = *(const v2f*)(Ar0 + ko);
      aA1 = *(const v2f*)(Ar1 + ko);
      bA0 = *(const v2f*)(Wr0 + ko);
      bA1 = *(const v2f*)(Wr1 + ko);
      bA2 = *(const v2f*)(Wr2 + ko);
      bA3 = *(const v2f*)(Wr3 + ko);
    }
    acc[0] = wmma4(aB0, bB0, acc[0]);
    acc[1] = wmma4(aB1, bB0, acc[1]);
    acc[2] = wmma4(aB0, bB1, acc[2]);
    acc[3] = wmma4(aB1, bB1, acc[3]);
    acc[4] = wmma4(aB0, bB2, acc[4]);
    acc[5] = wmma4(aB1, bB2, acc[5]);
    acc[6] = wmma4(aB0, bB3, acc[6]);
    acc[7] = wmma4(aB1, bB3, acc[7]);
  }

#pragma unroll
  for (int j = 0; j < 4; ++j) {
    const int col = n0 + j * 16 + lr;
    const float bv = bias[col];
#pragma unroll
    for (int mt = 0; mt < 2; ++mt) {
#pragma unroll
      for (int r = 0; r < 8; ++r) {
        const int row = m0 + mt * 16 + r + 8 * hi;
        float v = acc[2 * j + mt][r] + bv;
        if (mode == 1) v = R[(size_t)row * N + col] + 1.f / (1.f + expf(-v));
        C[(size_t)row * N + col] = v;
      }
    }
  }
}

// ---------------------------------------------------------------------------
// Flash attention, f32 WMMA. O = Q + softmax(Q K^T * 1/16) V
// Layout: Qp/Kp/Vp are (B*S, 256), head h owns cols [h*64, h*64+64).
// Wave: 16 queries; block: 128 queries; grid (S/128, B*H).
// ---------------------------------------------------------------------------
__global__ __launch_bounds__(256) void attn_flash_kernel(
    const float* __restrict__ Qp, const float* __restrict__ Kp,
    const float* __restrict__ Vp, float* __restrict__ O, int Sq, int Sk, int D,
    int H) {
  const float scale = 0.0625f;  // 1/sqrt(256)
  const int lane = threadIdx.x & 31;
  const int wv = threadIdx.x >> 5;
  const int lr = lane & 15;
  const int hi = lane >> 4;
  const int bh = blockIdx.y;
  const int b = bh / H, h = bh % H;
  const int q0 = blockIdx.x * 128 + wv * 16;

  const float* Qrow = Qp + ((size_t)(b * Sq + q0 + lr)) * D + h * 64;
  v2f aq[16];
#pragma unroll
  for (int k4 = 0; k4 < 16; ++k4) aq[k4] = *(const v2f*)(Qrow + k4 * 4 + 2 * hi);

  v8f accO[4];
#pragma unroll
  for (int j = 0; j < 4; ++j) accO[j] = (v8f){};
  float mrow[8], lrow[8];
#pragma unroll
  for (int r = 0; r < 8; ++r) {
    mrow[r] = -3.0e38f;
    lrow[r] = 0.f;
  }

  __shared__ float pls[8][16 * 17];  // per-wave P transpose tile, 17-stride pad
  float* myp = pls[wv];

  const float* Kbase = Kp + ((size_t)b * Sk) * D + h * 64;
  const float* Vbase = Vp + ((size_t)b * Sk) * D + h * 64;

  for (int n0 = 0; n0 < Sk; n0 += 16) {
    // S_tile = Q(16x64) @ K_tile^T(64x16); K-frags batched in groups of 8 so
    // loads are in flight ahead of the consuming WMMAs.
    v8f s = {};
    const float* Krow = Kbase + (size_t)(n0 + lr) * D;
    {
      v2f bk[8];
#pragma unroll
      for (int k4 = 0; k4 < 8; ++k4)
        bk[k4] = *(const v2f*)(Krow + k4 * 4 + 2 * hi);
#pragma unroll
      for (int k4 = 0; k4 < 8; ++k4) s = wmma4(aq[k4], bk[k4], s);
#pragma unroll
      for (int k4 = 0; k4 < 8; ++k4)
        bk[k4] = *(const v2f*)(Krow + (k4 + 8) * 4 + 2 * hi);
#pragma unroll
      for (int k4 = 0; k4 < 8; ++k4) s = wmma4(aq[k4 + 8], bk[k4], s);
    }

    // Issue all V-tile loads now; their latency overlaps the softmax VALU work
    // below, and they are consumed only after the LDS transpose.
    float vfx[4][4], vfy[4][4];  // [k4][j]
#pragma unroll
    for (int k4 = 0; k4 < 4; ++k4) {
      const int ko = k4 * 4 + 2 * hi;
      const float* Vr0 = Vbase + (size_t)(n0 + ko) * D;
      const float* Vr1 = Vr0 + D;
#pragma unroll
      for (int j = 0; j < 4; ++j) {
        vfx[k4][j] = Vr0[j * 16 + lr];
        vfy[k4][j] = Vr1[j * 16 + lr];
      }
    }

    // online softmax update (rows owned by half-wave: lanes 0-15 -> r, 16-31 -> r+8)
    float f[8];
#pragma unroll
    for (int r = 0; r < 8; ++r) {
      float sv = s[r] * scale;
      float mx = sv;
      mx = fmaxf(mx, __shfl_xor(mx, 1, 32));
      mx = fmaxf(mx, __shfl_xor(mx, 2, 32));
      mx = fmaxf(mx, __shfl_xor(mx, 4, 32));
      mx = fmaxf(mx, __shfl_xor(mx, 8, 32));
      float mnew = fmaxf(mrow[r], mx);
      float fe = expf(mrow[r] - mnew);
      float p = expf(sv - mnew);
      float ps = p;
      ps += __shfl_xor(ps, 1, 32);
      ps += __shfl_xor(ps, 2, 32);
      ps += __shfl_xor(ps, 4, 32);
      ps += __shfl_xor(ps, 8, 32);
      lrow[r] = lrow[r] * fe + ps;
      mrow[r] = mnew;
      f[r] = fe;
      s[r] = p;  // reuse s as P tile
    }
#pragma unroll
    for (int j = 0; j < 4; ++j)
#pragma unroll
      for (int r = 0; r < 8; ++r) accO[j][r] *= f[r];

    // transpose P through LDS: C-layout -> A-layout
    __syncthreads();
#pragma unroll
    for (int r = 0; r < 8; ++r) myp[(r + 8 * hi) * 17 + lr] = s[r];
    __syncthreads();

    // accO += P(16x16) @ V_tile(16x64)
#pragma unroll
    for (int k4 = 0; k4 < 4; ++k4) {
      const int ko = k4 * 4 + 2 * hi;
      v2f ap;
      ap.x = myp[lr * 17 + ko];
      ap.y = myp[lr * 17 + ko + 1];
#pragma unroll
      for (int j = 0; j < 4; ++j) {
        v2f bv;
        bv.x = vfx[k4][j];
        bv.y = vfy[k4][j];
        accO[j] = wmma4(ap, bv, accO[j]);
      }
    }
  }

  // O = Q + accO / l
#pragma unroll
  for (int j = 0; j < 4; ++j) {
#pragma unroll
    for (int r = 0; r < 8; ++r) {
      const int row = q0 + r + 8 * hi;
      const int col = h * 64 + j * 16 + lr;
      const size_t idx = (size_t)(b * Sq + row) * D + col;
      O[idx] = Qp[idx] + accO[j][r] / lrow[r];
    }
  }
}

// ---------------------------------------------------------------------------
// LayerNorm (last dim = 256), wave per row, in-place
// ---------------------------------------------------------------------------
__global__ __launch_bounds__(256) void ln_kernel(float* __restrict__ X) {
  const int lane = threadIdx.x & 31;
  const int wv = threadIdx.x >> 5;
  const int row = blockIdx.x * 8 + wv;
  float* p = X + (size_t)row * 256;
  const int c0 = lane * 8;
  float v[8];
#pragma unroll
  for (int i = 0; i < 8; ++i) v[i] = p[c0 + i];
  float s = 0.f;
#pragma unroll
  for (int i = 0; i < 8; ++i) s += v[i];
  const float mean = wave_sum32(s) * (1.0f / 256.0f);
  float q = 0.f;
#pragma unroll
  for (int i = 0; i < 8; ++i) {
    float d = v[i] - mean;
    q += d * d;
  }
  const float rs = rsqrtf(wave_sum32(q) * (1.0f / 256.0f) + 1e-5f);
#pragma unroll
  for (int i = 0; i < 8; ++i) p[c0 + i] = (v[i] - mean) * rs;
}

__global__ __launch_bounds__(256) void sigmoid_kernel(float* __restrict__ Z) {
  const int idx = blockIdx.x * 256 + threadIdx.x;
  Z[idx] = 1.f / (1.f + expf(-Z[idx]));
}

// ---------------------------------------------------------------------------
// PMA tail (1 query). Tiny — plain VALU kernels.
// ---------------------------------------------------------------------------
__global__ __launch_bounds__(256) void pma_qvec_kernel(
    const float* __restrict__ S, const float* __restrict__ Wq,
    const float* __restrict__ bq, float* __restrict__ qvec) {
  const int n = threadIdx.x;  // 256
  float a = bq[n];
  for (int k = 0; k < 256; ++k) a += S[k] * Wq[(size_t)n * 256 + k];
  qvec[n] = a;
}

__global__ __launch_bounds__(512) void pma_attn_kernel(
    const float* __restrict__ qvec, const float* __restrict__ Kp,
    const float* __restrict__ Vp, float* __restrict__ Osm) {
  const int t = threadIdx.x;  // 512 keys
  const int b = blockIdx.x >> 2, h = blockIdx.x & 3;
  __shared__ float qs[64], sc[512], red[512];
  if (t < 64) qs[t] = qvec[h * 64 + t];
  __syncthreads();
  const float* kr = Kp + ((size_t)(b * 512 + t)) * 256 + h * 64;
  float s = 0.f;
  for (int d = 0; d < 64; ++d) s += qs[d] * kr[d];
  s *= 0.0625f;
  sc[t] = s;
  red[t] = s;
  __syncthreads();
  for (int st = 256; st > 0; st >>= 1) {
    if (t < st) red[t] = fmaxf(red[t], red[t + st]);
    __syncthreads();
  }
  const float mx = red[0];
  __syncthreads();
  const float pv = expf(s - mx);
  sc[t] = pv;
  red[t] = pv;
  __syncthreads();
  for (int st = 256; st > 0; st >>= 1) {
    if (t < st) red[t] += red[t + st];
    __syncthreads();
  }
  const float l = red[0];
  __syncthreads();
  if (t < 64) {
    float o = 0.f;
    for (int k = 0; k < 512; ++k)
      o += sc[k] * Vp[((size_t)(b * 512 + k)) * 256 + h * 64 + t];
    Osm[b * 256 + h * 64 + t] = qs[t] + o / l;
  }
}

__global__ __launch_bounds__(256) void pma_post_kernel(
    const float* __restrict__ Osm, const float* __restrict__ Wo,
    const float* __restrict__ bo, const float* __restrict__ outw,
    const float* __restrict__ outb, float* __restrict__ out) {
  const int t = threadIdx.x, b = blockIdx.x;
  __shared__ float buf[256], red[256];
  float x = Osm[b * 256 + t];
  // LN1
  red[t] = x;
  __syncthreads();
  for (int st = 128; st > 0; st >>= 1) {
    if (t < st) red[t] += red[t + st];
    __syncthreads();
  }
  const float mean = red[0] * (1.0f / 256.0f);
  __syncthreads();
  float d = x - mean;
  red[t] = d * d;
  __syncthreads();
  for (int st = 128; st > 0; st >>= 1) {
    if (t < st) red[t] += red[t + st];
    __syncthreads();
  }
  x = d * rsqrtf(red[0] * (1.0f / 256.0f) + 1e-5f);
  __syncthreads();
  buf[t] = x;
  __syncthreads();
  float a = bo[t];
  for (int k = 0; k < 256; ++k) a += buf[k] * Wo[(size_t)t * 256 + k];
  const float u = x + 1.f / (1.f + expf(-a));
  // LN2
  red[t] = u;
  __syncthreads();
  for (int st = 128; st > 0; st >>= 1) {
    if (t < st) red[t] += red[t + st];
    __syncthreads();
  }
  const float mean2 = red[0] * (1.0f / 256.0f);
  __syncthreads();
  const float d2 = u - mean2;
  red[t] = d2 * d2;
  __syncthreads();
  for (int st = 128; st > 0; st >>= 1) {
    if (t < st) red[t] += red[t + st];
    __syncthreads();
  }
  const float z = d2 * rsqrtf(red[0] * (1.0f / 256.0f) + 1e-5f);
  __syncthreads();
  red[t] = z * outw[t];
  __syncthreads();
  for (int st = 128; st > 0; st >>= 1) {
    if (t < st) red[t] += red[t + st];
    __syncthreads();
  }
  if (t == 0) out[b] = 1.f / (1.f + expf(-(red[0] + outb[0])));
}

// ---------------------------------------------------------------------------
extern "C" void kernel_launch(void* const* d_in, const int* in_sizes, int n_in,
                              void* d_out, int out_size, void* d_ws,
                              size_t ws_size, hipStream_t stream) {
  (void)in_sizes;
  (void)n_in;
  (void)out_size;
  (void)ws_size;
  const float* X = (const float*)d_in[0];
  const float* conv_w = (const float*)d_in[1];
  const float* bn_g = (const float*)d_in[2];
  const float* bn_b = (const float*)d_in[3];
  const float* s1_wq = (const float*)d_in[4];
  const float* s1_bq = (const float*)d_in[5];
  const float* s1_wk = (const float*)d_in[6];
  const float* s1_bk = (const float*)d_in[7];
  const float* s1_wv = (const float*)d_in[8];
  const float* s1_bv = (const float*)d_in[9];
  const float* s1_wo = (const float*)d_in[10];
  const float* s1_bo = (const float*)d_in[11];
  const float* s23_wq = (const float*)d_in[12];
  const float* s23_bq = (const float*)d_in[13];
  const float* s23_wk = (const float*)d_in[14];
  const float* s23_bk = (const float*)d_in[15];
  const float* s23_wv = (const float*)d_in[16];
  const float* s23_bv = (const float*)d_in[17];
  const float* s23_wo = (const float*)d_in[18];
  const float* s23_bo = (const float*)d_in[19];
  const float* pma_S = (const float*)d_in[20];
  const float* p_wq = (const float*)d_in[21];
  const float* p_bq = (const float*)d_in[22];
  const float* p_wk = (const float*)d_in[23];
  const float* p_bk = (const float*)d_in[24];
  const float* p_wv = (const float*)d_in[25];
  const float* p_bv = (const float*)d_in[26];
  const float* p_wo = (const float*)d_in[27];
  const float* p_bo = (const float*)d_in[28];
  const float* out_w = (const float*)d_in[29];
  const float* out_b = (const float*)d_in[30];
  float* out = (float*)d_out;

  float* ws = (float*)d_ws;
  const size_t SLOT = (size_t)32768 * 288;
  float* gsum = ws;
  float* gsq = ws + 32;
  float* scalev = ws + 64;
  float* shiftv = ws + 96;
  float* qvec = ws + 128;
  float* Osm = ws + 1024;  // 64*256
  float* buf0 = ws + 32768;
  float* buf1 = buf0 + SLOT;
  float* buf2 = buf1 + SLOT;
  float* buf3 = buf2 + SLOT;
  float* buf4 = buf3 + SLOT;
  float* buf5 = buf4 + SLOT;

  const int M = 32768;
  const dim3 gemmGrid(M / 256, 4);
  const dim3 attnGrid(512 / 128, 64 * 4);

  // conv + batchnorm
  (void)hipMemsetAsync(gsum, 0, 64 * sizeof(float), stream);
  conv_stats_kernel<<<4096, 256, 0, stream>>>(X, conv_w, buf1, gsum, gsq);
  bn_finalize_kernel<<<1, 32, 0, stream>>>(gsum, gsq, bn_g, bn_b, scalev,
                                           shiftv);
  bn_apply_kernel<<<36864, 256, 0, stream>>>(buf1, scalev, shiftv, buf0);

  // SAB1 (Din = 288)
  gemm_wmma_kernel<<<gemmGrid, 256, 0, stream>>>(buf0, s1_wq, s1_bq, nullptr,
                                                 buf1, M, 256, 288, 0);
  gemm_wmma_kernel<<<gemmGrid, 256, 0, stream>>>(buf0, s1_wk, s1_bk, nullptr,
                                                 buf2, M, 256, 288, 0);
  gemm_wmma_kernel<<<gemmGrid, 256, 0, stream>>>(buf0, s1_wv, s1_bv, nullptr,
                                                 buf3, M, 256, 288, 0);
  attn_flash_kernel<<<attnGrid, 256, 0, stream>>>(buf1, buf2, buf3, buf4, 512,
                                                  512, 256, 4);
  ln_kernel<<<4096, 256, 0, stream>>>(buf4);
  gemm_wmma_kernel<<<gemmGrid, 256, 0, stream>>>(buf4, s1_wo, s1_bo, buf4, buf5,
                                                 M, 256, 256, 1);
  ln_kernel<<<4096, 256, 0, stream>>>(buf5);

  // SAB2, SAB3 (Din = 256), ping-pong buf5 <-> buf0
  float* zin = buf5;
  float* zout = buf0;
  for (int i = 0; i < 2; ++i) {
    const float* wq = s23_wq + (size_t)i * 65536;
    const float* bq = s23_bq + i * 256;
    const float* wk = s23_wk + (size_t)i * 65536;
    const float* bk = s23_bk + i * 256;
    const float* wv = s23_wv + (size_t)i * 65536;
    const float* bv = s23_bv + i * 256;
    const float* wo = s23_wo + (size_t)i * 65536;
    const float* bo = s23_bo + i * 256;
    gemm_wmma_kernel<<<gemmGrid, 256, 0, stream>>>(zin, wq, bq, nullptr, buf1,
                                                   M, 256, 256, 0);
    gemm_wmma_kernel<<<gemmGrid, 256, 0, stream>>>(zin, wk, bk, nullptr, buf2,
                                                   M, 256, 256, 0);
    gemm_wmma_kernel<<<gemmGrid, 256, 0, stream>>>(zin, wv, bv, nullptr, buf3,
                                                   M, 256, 256, 0);
    attn_flash_kernel<<<attnGrid, 256, 0, stream>>>(buf1, buf2, buf3, buf4, 512,
                                                    512, 256, 4);
    ln_kernel<<<4096, 256, 0, stream>>>(buf4);
    gemm_wmma_kernel<<<gemmGrid, 256, 0, stream>>>(buf4, wo, bo, buf4, zout, M,
                                                   256, 256, 1);
    ln_kernel<<<4096, 256, 0, stream>>>(zout);
    float* t = zin;
    zin = zout;
    zout = t;
  }
  // final z in buf5 (zin). Apply sigmoid.
  sigmoid_kernel<<<32768, 256, 0, stream>>>(buf5);

  // PMA: K-side projections via WMMA GEMMs, tiny Q-side and attention on VALU.
  pma_qvec_kernel<<<1, 256, 0, stream>>>(pma_S, p_wq, p_bq, qvec);
  gemm_wmma_kernel<<<gemmGrid, 256, 0, stream>>>(buf5, p_wk, p_bk, nullptr,
                                                 buf2, M, 256, 256, 0);
  gemm_wmma_kernel<<<gemmGrid, 256, 0, stream>>>(buf5, p_wv, p_bv, nullptr,
                                                 buf3, M, 256, 256, 0);
  pma_attn_kernel<<<256, 512, 0, stream>>>(qvec, buf2, buf3, Osm);
  pma_post_kernel<<<64, 256, 0, stream>>>(Osm, p_wo, p_bo, out_w, out_b, out);
}